// LocalAggregationLossModule_42795054137893
// MI455X (gfx1250) — compile-verified
//
#include <hip/hip_runtime.h>
#include <math.h>

typedef float v2f __attribute__((ext_vector_type(2)));
typedef float v8f __attribute__((ext_vector_type(8)));
typedef unsigned int u32x4 __attribute__((ext_vector_type(4)));
typedef int i32x4 __attribute__((ext_vector_type(4)));
typedef int i32x8 __attribute__((ext_vector_type(8)));

#define BATCH 256
#define DIM   128
#define NKM   3
#define KTOP  4096
#define NBINS 2048
#define CHUNK 128
#define LSTR  (DIM + 4)     // LDS row stride (floats): TDM pads 4 DWORDs / 128 DWORDs
#define BCAP  2048          // boundary-bin candidate capacity per row
#define TINV  (1.0f / 0.07f)

// ---------------------------------------------------------------- utilities
__global__ void zero_kernel(unsigned int* __restrict__ p, size_t n) {
  size_t i = (size_t)blockIdx.x * blockDim.x + threadIdx.x;
  size_t stride = (size_t)gridDim.x * blockDim.x;
  for (; i < n; i += stride) p[i] = 0u;
}

// normalize outputs -> out_n ; gather batch labels
__global__ void prep_kernel(const float* __restrict__ outputs,
                            const int* __restrict__ indices,
                            const int* __restrict__ clab,
                            float* __restrict__ out_n,
                            int* __restrict__ blab, int N) {
  __shared__ float s[DIM];
  int b = blockIdx.x, t = threadIdx.x;
  float v = outputs[b * DIM + t];
  s[t] = v * v;
  __syncthreads();
  for (int m = DIM / 2; m > 0; m >>= 1) {
    if (t < m) s[t] += s[t + m];
    __syncthreads();
  }
  float inv = rsqrtf(s[0]);
  out_n[b * DIM + t] = v * inv;
  if (t < NKM) blab[t * BATCH + b] = clab[(size_t)t * N + indices[b]];
}

// ------------------------------------------------- fused WMMA GEMM core
// MODE 0: histogram pass      MODE 1: accumulate pass
// 512 threads = 16 waves; wave w owns M-tile w (batch rows 16w..16w+15).
template <int MODE>
__global__ __launch_bounds__(512, 1)
void gemm_kernel(const float* __restrict__ out_n,
                 const float* __restrict__ mb,
                 const int* __restrict__ clab,
                 const int* __restrict__ blab,
                 const int* __restrict__ rowinfo,
                 int* __restrict__ hist,
                 float* __restrict__ denp,
                 float* __restrict__ nump,
                 int* __restrict__ bcnt,
                 float* __restrict__ bval,
                 int* __restrict__ bidx,
                 int N, int nch) {
  extern __shared__ float smem[];  // CHUNK * LSTR floats, filled by TDM
  const int chunk = blockIdx.x;
  const int j0 = chunk * CHUNK;
  const int tid = threadIdx.x;

  // ---- Tensor Data Mover: stage 128x128 f32 tile of memory_bank into LDS.
  // 2D descriptor; tensor_dim1 = N - j0 so the TDM zero-fills OOB rows of the
  // last chunk; pad_enable inserts 4 DWORDs after every 128 DWORDs, giving the
  // bank-conflict-free LSTR=132 row stride directly in hardware.
  if (tid < 32) {
    unsigned long long gaddr = (unsigned long long)(uintptr_t)(mb + (size_t)j0 * DIM);
    unsigned int lds_addr = (unsigned int)(uintptr_t)(&smem[0]);  // LDS byte offset
    unsigned int rows = (unsigned int)(N - j0);
    u32x4 g0;
    g0.x = 1u;                                                  // count=1, user mode
    g0.y = lds_addr;                                            // lds_addr [63:32]
    g0.z = (unsigned int)(gaddr & 0xFFFFFFFFull);               // global_addr lo
    g0.w = (unsigned int)((gaddr >> 32) & 0x1FFFFFFull)         // global_addr [56:32]
         | (2u << 30);                                          // type = 2 (image)
    i32x8 g1;
    g1[0] = (int)((2u << 16)      // data_size = 4B
                | (1u << 20)      // pad_enable
                | (6u << 22)      // pad_interval: 128 DWORDs
                | (3u << 25));    // pad_amount:   4 DWORDs
    g1[1] = (int)((unsigned)DIM << 16);                         // tensor_dim0 lo16
    g1[2] = (int)((rows & 0xFFFFu) << 16);                      // tensor_dim1 lo16
    g1[3] = (int)(((rows >> 16) & 0xFFFFu)                      // tensor_dim1 hi16
                | ((unsigned)DIM << 16));                       // tile_dim0 = 128
    g1[4] = (int)(unsigned)CHUNK;                               // tile_dim1 = 128
    g1[5] = (int)(unsigned)DIM;                                 // tensor_dim0_stride lo32
    g1[6] = 0;                                                  // stride hi / dim1_stride lo
    g1[7] = 0;
    i32x4 gz4 = {0, 0, 0, 0};
    i32x8 gz8 = {0, 0, 0, 0, 0, 0, 0, 0};
    __builtin_amdgcn_tensor_load_to_lds(g0, g1, gz4, gz4, gz8, 0);
    __builtin_amdgcn_s_wait_tensorcnt((short)0);
  }
  __syncthreads();

  const int lane = tid & 31;
  const int wave = tid >> 5;    // 0..15 = M-tile
  const int half = lane >> 4;   // K-sublane selector for 16x16x4 layout
  const int l16  = lane & 15;

  // A fragments: 1 M-tile x 32 k-steps, persistent in VGPRs (64 VGPRs).
  // Lanes 0-15 hold K = 4k+0,4k+1 ; lanes 16-31 hold K = 4k+2,4k+3.
  v2f afrag[32];
  {
    int row = wave * 16 + l16;
    const float* ap = out_n + row * DIM + half * 2;
#pragma unroll
    for (int ks = 0; ks < 32; ++ks) afrag[ks] = *(const v2f*)(ap + ks * 4);
  }

  float dacc[8], nacc[8];
  if (MODE == 1) {
#pragma unroll
    for (int v = 0; v < 8; ++v) { dacc[v] = 0.f; nacc[v] = 0.f; }
  }

  for (int g = 0; g < 8; ++g) {  // 8 groups of 16 bank entries per chunk
    v8f acc = {0, 0, 0, 0, 0, 0, 0, 0};
    const float* bp = &smem[(g * 16 + l16) * LSTR + half * 2];
#pragma unroll
    for (int ks = 0; ks < 32; ++ks) {
      v2f bf = *(const v2f*)(bp + ks * 4);
      acc = __builtin_amdgcn_wmma_f32_16x16x4_f32(false, afrag[ks], false, bf,
                                                  (short)0, acc, false, false);
    }
    const int j = j0 + g * 16 + l16;   // bank entry index (column)
#pragma unroll
    for (int v = 0; v < 8; ++v) {
      float d = acc[v];
      int r = wave * 16 + v + half * 8;  // batch row (C layout)
      if (j < N) {
        int bin = (int)((d + 1.0f) * (NBINS * 0.5f));
        bin = bin < 0 ? 0 : (bin > NBINS - 1 ? NBINS - 1 : bin);
        if (MODE == 0) {
          atomicAdd(&hist[(size_t)r * NBINS + bin], 1);
        } else {
          int tb = rowinfo[2 * r];
          if (bin > tb) {
            float p = __expf(d * TINV);
            bool match = false;
#pragma unroll
            for (int k = 0; k < NKM; ++k)
              match = match || (clab[(size_t)k * N + j] == blab[k * BATCH + r]);
            dacc[v] += p;
            if (match) nacc[v] += p;
          } else if (bin == tb) {
            int pos = atomicAdd(&bcnt[r], 1);
            if (pos < BCAP) {
              bval[(size_t)r * BCAP + pos] = d;
              bidx[(size_t)r * BCAP + pos] = j;
            }
          }
        }
      }
    }
  }

  if (MODE == 1) {
    // deterministic per-(row,chunk) partials: butterfly-sum the 16 N-lanes
#pragma unroll
    for (int v = 0; v < 8; ++v) {
      float ds = dacc[v], ns = nacc[v];
      for (int m = 8; m > 0; m >>= 1) {
        ds += __shfl_xor(ds, m, 16);
        ns += __shfl_xor(ns, m, 16);
      }
      if (l16 == 0) {
        int r = wave * 16 + v + half * 8;
        denp[(size_t)r * nch + chunk] = ds;
        nump[(size_t)r * nch + chunk] = ns;
      }
    }
  }
}

// per-row threshold bin: largest bin tb with count(bins >= tb) >= KTOP
__global__ void scan_kernel(const int* __restrict__ hist, int* __restrict__ rowinfo) {
  int r = blockIdx.x;
  if (threadIdx.x != 0) return;
  int cum = 0, tb = 0, rn = KTOP;
  for (int b2 = NBINS - 1; b2 >= 0; --b2) {
    int c = hist[(size_t)r * NBINS + b2];
    if (cum + c >= KTOP) { tb = b2; rn = KTOP - cum; break; }
    cum += c;
    tb = b2; rn = KTOP - cum;
  }
  rowinfo[2 * r] = tb;
  rowinfo[2 * r + 1] = rn;
}

// fixed-order reduction of per-chunk partials
__global__ void reduce_parts_kernel(const float* __restrict__ denp,
                                    const float* __restrict__ nump,
                                    float* __restrict__ den, float* __restrict__ num,
                                    int nch) {
  __shared__ float sd[256], sn[256];
  int r = blockIdx.x, t = threadIdx.x;
  float ds = 0.f, ns = 0.f;
  for (int c = t; c < nch; c += 256) {
    ds += denp[(size_t)r * nch + c];
    ns += nump[(size_t)r * nch + c];
  }
  sd[t] = ds; sn[t] = ns;
  __syncthreads();
  for (int m = 128; m > 0; m >>= 1) {
    if (t < m) { sd[t] += sd[t + m]; sn[t] += sn[t + m]; }
    __syncthreads();
  }
  if (t == 0) { den[r] = sd[0]; num[r] = sn[0]; }
}

// exact rank-select of the remaining r elements from the boundary bin
__global__ void bnd_kernel(const float* __restrict__ bval, const int* __restrict__ bidx,
                           const int* __restrict__ bcnt, const int* __restrict__ rowinfo,
                           const int* __restrict__ clab, const int* __restrict__ blab,
                           float* __restrict__ den, float* __restrict__ num, int N) {
  __shared__ float sd[256], sn[256];
  int r = blockIdx.x, t = threadIdx.x;
  int cnt = bcnt[r]; if (cnt > BCAP) cnt = BCAP;
  int rn = rowinfo[2 * r + 1];
  float ds = 0.f, ns = 0.f;
  for (int i = t; i < cnt; i += 256) {
    float v = bval[(size_t)r * BCAP + i];
    int j = bidx[(size_t)r * BCAP + i];
    int rank = 0;
    for (int q = 0; q < cnt; ++q) {
      float vq = bval[(size_t)r * BCAP + q];
      int jq = bidx[(size_t)r * BCAP + q];
      rank += (vq > v) || (vq == v && jq < j);
    }
    if (rank < rn) {
      float p = __expf(v * TINV);
      bool match = false;
      for (int k = 0; k < NKM; ++k)
        match = match || (clab[(size_t)k * N + j] == blab[k * BATCH + r]);
      ds += p;
      if (match) ns += p;
    }
  }
  sd[t] = ds; sn[t] = ns;
  __syncthreads();
  for (int m = 128; m > 0; m >>= 1) {
    if (t < m) { sd[t] += sd[t + m]; sn[t] += sn[t + m]; }
    __syncthreads();
  }
  if (t == 0) { den[r] += sd[0]; num[r] += sn[0]; }
}

// loss = -mean(log(numer/denom + 1e-7))   (Z cancels in the ratio)
__global__ void loss_kernel(const float* __restrict__ den, const float* __restrict__ num,
                            float* __restrict__ out) {
  __shared__ float s[BATCH];
  int t = threadIdx.x;
  float rp = num[t] / den[t];
  s[t] = logf(rp + 1e-7f);
  __syncthreads();
  for (int m = BATCH / 2; m > 0; m >>= 1) {
    if (t < m) s[t] += s[t + m];
    __syncthreads();
  }
  if (t == 0) out[0] = -s[0] / (float)BATCH;
}

// new_data_memory = l2norm(0.5*mb[idx] + 0.5*out_n)
__global__ void newmem_kernel(const int* __restrict__ indices,
                              const float* __restrict__ out_n,
                              const float* __restrict__ mb,
                              float* __restrict__ out) {
  __shared__ float s[DIM];
  int b = blockIdx.x, t = threadIdx.x;
  int j = indices[b];
  float v = 0.5f * mb[(size_t)j * DIM + t] + 0.5f * out_n[b * DIM + t];
  s[t] = v * v;
  __syncthreads();
  for (int m = DIM / 2; m > 0; m >>= 1) {
    if (t < m) s[t] += s[t + m];
    __syncthreads();
  }
  out[1 + b * DIM + t] = v * rsqrtf(s[0]);
}

// ---------------------------------------------------------------- launcher
extern "C" void kernel_launch(void* const* d_in, const int* in_sizes, int n_in,
                              void* d_out, int out_size, void* d_ws, size_t ws_size,
                              hipStream_t stream) {
  const int*   indices = (const int*)d_in[0];
  const float* outputs = (const float*)d_in[1];
  const float* mb      = (const float*)d_in[2];
  const int*   clab    = (const int*)d_in[3];
  float* out = (float*)d_out;
  const int N   = in_sizes[2] / DIM;             // 200000
  const int nch = (N + CHUNK - 1) / CHUNK;       // 1563

  char* ws = (char*)d_ws;
  size_t off = 0;
  auto take = [&](size_t bytes) -> void* {
    void* p = ws + off;
    off += (bytes + 255) & ~(size_t)255;
    return p;
  };
  float* out_n   = (float*)take((size_t)BATCH * DIM * 4);
  int*   blab    = (int*)  take((size_t)NKM * BATCH * 4);
  int*   hist    = (int*)  take((size_t)BATCH * NBINS * 4);
  int*   rowinfo = (int*)  take((size_t)BATCH * 2 * 4);
  float* denp    = (float*)take((size_t)BATCH * nch * 4);
  float* nump    = (float*)take((size_t)BATCH * nch * 4);
  float* den     = (float*)take((size_t)BATCH * 4);
  float* num     = (float*)take((size_t)BATCH * 4);
  int*   bcnt    = (int*)  take((size_t)BATCH * 4);
  float* bval    = (float*)take((size_t)BATCH * BCAP * 4);
  int*   bidx    = (int*)  take((size_t)BATCH * BCAP * 4);
  (void)ws_size; (void)n_in; (void)out_size;

  zero_kernel<<<512, 256, 0, stream>>>((unsigned int*)hist, (size_t)BATCH * NBINS);
  zero_kernel<<<1, 256, 0, stream>>>((unsigned int*)bcnt, (size_t)BATCH);
  prep_kernel<<<BATCH, DIM, 0, stream>>>(outputs, indices, clab, out_n, blab, N);

  size_t smem = (size_t)CHUNK * LSTR * sizeof(float);
  gemm_kernel<0><<<nch, 512, smem, stream>>>(out_n, mb, nullptr, nullptr, nullptr,
                                             hist, nullptr, nullptr, nullptr,
                                             nullptr, nullptr, N, nch);
  scan_kernel<<<BATCH, 32, 0, stream>>>(hist, rowinfo);
  gemm_kernel<1><<<nch, 512, smem, stream>>>(out_n, mb, clab, blab, rowinfo,
                                             nullptr, denp, nump, bcnt,
                                             bval, bidx, N, nch);
  reduce_parts_kernel<<<BATCH, 256, 0, stream>>>(denp, nump, den, num, nch);
  bnd_kernel<<<BATCH, 256, 0, stream>>>(bval, bidx, bcnt, rowinfo, clab, blab, den, num, N);
  loss_kernel<<<1, BATCH, 0, stream>>>(den, num, out);
  newmem_kernel<<<BATCH, DIM, 0, stream>>>(indices, out_n, mb, out);
}